// KernelRegressor_8126078124621
// MI455X (gfx1250) — compile-verified
//
#include <hip/hip_runtime.h>
#include <hip/hip_bf16.h>

typedef __attribute__((ext_vector_type(2))) float v2f;
typedef __attribute__((ext_vector_type(8))) float v8f;

#define N_NODES 1024
#define IN_FEAT 128
#define N_HEADS 4
#define N_HID   32
#define SLOPE   0.2f

// ---------------------------------------------------------------------------
// Kernel 1: g = h @ W   [1024x128] @ [128x128] -> [1024x128]
// One wave32 per 16x16 output tile, V_WMMA_F32_16X16X4_F32, K-loop of 32 steps.
// A layout (16x4 f32): lanes 0-15 hold (M=lane, K=0/1), lanes 16-31 (M=lane-16, K=2/3).
// B layout (4x16 f32): mirrored: lanes 0-15 hold (K=0/1, N=lane), lanes 16-31 (K=2/3, N=lane-16).
// C/D: VGPR r -> (M=r, N=lane) for lanes 0-15, (M=r+8, N=lane-16) for lanes 16-31.
// ---------------------------------------------------------------------------
__global__ __launch_bounds__(32) void gat_proj_wmma(const float* __restrict__ h,
                                                    const float* __restrict__ W,
                                                    float* __restrict__ g) {
  const int m0   = blockIdx.x * 16;
  const int n0   = blockIdx.y * 16;
  const int lane = threadIdx.x;
  const int half = lane >> 4;     // 0: K=0,1   1: K=2,3
  const int l15  = lane & 15;

  v8f c = {};
#pragma unroll 4
  for (int k = 0; k < IN_FEAT; k += 4) {
    v2f a, b;
    const int ka = k + 2 * half;
    a.x = h[(m0 + l15) * IN_FEAT + ka + 0];
    a.y = h[(m0 + l15) * IN_FEAT + ka + 1];
    b.x = W[(ka + 0) * (N_HEADS * N_HID) + n0 + l15];
    b.y = W[(ka + 1) * (N_HEADS * N_HID) + n0 + l15];
    c = __builtin_amdgcn_wmma_f32_16x16x4_f32(false, a, false, b, (short)0, c,
                                              false, false);
  }
#pragma unroll
  for (int r = 0; r < 8; ++r) {
    g[(m0 + r + 8 * half) * (N_HEADS * N_HID) + n0 + l15] = c[r];
  }
}

// ---------------------------------------------------------------------------
// Kernel 2: pairwise scores e[i,j,h] = a_w . lrelu(g[i]+g[j]), softmax over j,
// write att[h][i][j] (ws), raw a_mean row (into d_out), rowmax[i] (ws).
// One 256-thread block per destination node i; each thread owns 4 j's.
// ---------------------------------------------------------------------------
__global__ __launch_bounds__(256) void gat_scores(const float* __restrict__ g,
                                                  const float* __restrict__ a_w,
                                                  float* __restrict__ att,
                                                  float* __restrict__ a_raw,
                                                  float* __restrict__ rowmax) {
  const int i = blockIdx.x;
  const int t = threadIdx.x;

  __shared__ float gi[N_HEADS * N_HID];
  __shared__ float aw[N_HID];
  __shared__ float red[256];

  if (t < N_HEADS * N_HID) gi[t] = g[i * (N_HEADS * N_HID) + t];
  if (t < N_HID) aw[t] = a_w[t];
  __syncthreads();

  float e[4][N_HEADS];
#pragma unroll
  for (int jj = 0; jj < 4; ++jj) {
    const int j = t + jj * 256;
    const float* gj = g + j * (N_HEADS * N_HID);
#pragma unroll
    for (int hh = 0; hh < N_HEADS; ++hh) {
      float acc = 0.f;
#pragma unroll
      for (int f = 0; f < N_HID; ++f) {
        const float s = gi[hh * N_HID + f] + gj[hh * N_HID + f];
        const float l = (s >= 0.f) ? s : SLOPE * s;
        acc = fmaf(l, aw[f], acc);
      }
      e[jj][hh] = acc;
    }
  }

  float sm[N_HEADS];
#pragma unroll
  for (int hh = 0; hh < N_HEADS; ++hh) {
    // block max
    float lm = fmaxf(fmaxf(e[0][hh], e[1][hh]), fmaxf(e[2][hh], e[3][hh]));
    red[t] = lm;
    __syncthreads();
    for (int s = 128; s > 0; s >>= 1) {
      if (t < s) red[t] = fmaxf(red[t], red[t + s]);
      __syncthreads();
    }
    const float mx = red[0];
    __syncthreads();
    // exp + block sum
    float ls = 0.f;
#pragma unroll
    for (int jj = 0; jj < 4; ++jj) {
      e[jj][hh] = __expf(e[jj][hh] - mx);
      ls += e[jj][hh];
    }
    red[t] = ls;
    __syncthreads();
    for (int s = 128; s > 0; s >>= 1) {
      if (t < s) red[t] += red[t + s];
      __syncthreads();
    }
    sm[hh] = red[0];
    __syncthreads();
  }

  // write att rows, raw head-mean, and reduce row max of head-mean
  float lrm = 0.f;
#pragma unroll
  for (int jj = 0; jj < 4; ++jj) {
    const int j = t + jj * 256;
    float am = 0.f;
#pragma unroll
    for (int hh = 0; hh < N_HEADS; ++hh) {
      const float a = e[jj][hh] / sm[hh];
      att[((size_t)hh << 20) + ((size_t)i << 10) + j] = a;
      am += a;
    }
    am *= 0.25f;
    a_raw[((size_t)i << 10) + j] = am;
    lrm = fmaxf(lrm, am);
  }
  red[t] = lrm;
  __syncthreads();
  for (int s = 128; s > 0; s >>= 1) {
    if (t < s) red[t] = fmaxf(red[t], red[t + s]);
    __syncthreads();
  }
  if (t == 0) rowmax[i] = red[0];
}

// ---------------------------------------------------------------------------
// Kernel 3: hid[i,f] = 0.25 * sum_h sum_j att_h[i,j] * g[j, h*32+f]
// One wave32 per 16x16 tile of hid [1024x32]; loops heads x K=1024/4 WMMA steps,
// accumulating all heads into one f32 accumulator.
// ---------------------------------------------------------------------------
__global__ __launch_bounds__(32) void gat_agg_wmma(const float* __restrict__ att,
                                                   const float* __restrict__ g,
                                                   float* __restrict__ hid) {
  const int m0   = blockIdx.x * 16;  // over i (64 tiles)
  const int n0   = blockIdx.y * 16;  // over f (2 tiles)
  const int lane = threadIdx.x;
  const int half = lane >> 4;
  const int l15  = lane & 15;

  v8f c = {};
  for (int hh = 0; hh < N_HEADS; ++hh) {
    const float* A = att + ((size_t)hh << 20);
    const float* B = g + hh * N_HID;
#pragma unroll 4
    for (int k = 0; k < N_NODES; k += 4) {
      const int ka = k + 2 * half;
      v2f a, b;
      a.x = A[(size_t)(m0 + l15) * N_NODES + ka + 0];
      a.y = A[(size_t)(m0 + l15) * N_NODES + ka + 1];
      b.x = B[(size_t)(ka + 0) * (N_HEADS * N_HID) + n0 + l15];
      b.y = B[(size_t)(ka + 1) * (N_HEADS * N_HID) + n0 + l15];
      c = __builtin_amdgcn_wmma_f32_16x16x4_f32(false, a, false, b, (short)0, c,
                                                false, false);
    }
  }
#pragma unroll
  for (int r = 0; r < 8; ++r) {
    hid[(m0 + r + 8 * half) * N_HID + n0 + l15] = 0.25f * c[r];
  }
}

// ---------------------------------------------------------------------------
// Kernel 4: a_mean[i,j] = clip(a_raw[i,j] / (rowmax[j] + 1e-10), 0, 1) in place.
// ---------------------------------------------------------------------------
__global__ __launch_bounds__(256) void gat_norm(float* __restrict__ a_mean,
                                                const float* __restrict__ rowmax) {
  const int idx = blockIdx.x * blockDim.x + threadIdx.x;
  const int j = idx & (N_NODES - 1);
  const float v = a_mean[idx] / (rowmax[j] + 1e-10f);
  a_mean[idx] = fminf(fmaxf(v, 0.f), 1.f);
}

extern "C" void kernel_launch(void* const* d_in, const int* in_sizes, int n_in,
                              void* d_out, int out_size, void* d_ws, size_t ws_size,
                              hipStream_t stream) {
  const float* h   = (const float*)d_in[0];   // [1024,128]
  const float* W   = (const float*)d_in[1];   // [128,128]
  const float* a_w = (const float*)d_in[2];   // [32]

  float* hid    = (float*)d_out;                      // [1024,32]
  float* a_mean = (float*)d_out + N_NODES * N_HID;    // [1024,1024]

  char* ws = (char*)d_ws;
  float* g      = (float*)ws;                                      // 512 KB
  float* att    = (float*)(ws + (size_t)N_NODES * IN_FEAT * 4);    // 16 MB
  float* rowmax = (float*)(ws + (size_t)N_NODES * IN_FEAT * 4 +
                           (size_t)N_HEADS * N_NODES * N_NODES * 4);  // 4 KB

  // 1) projection g = h @ W  (WMMA f32)
  gat_proj_wmma<<<dim3(N_NODES / 16, (N_HEADS * N_HID) / 16), 32, 0, stream>>>(h, W, g);

  // 2) pairwise scores + softmax + raw a_mean + row maxima
  gat_scores<<<N_NODES, 256, 0, stream>>>(g, a_w, att, a_mean, rowmax);

  // 3) attention-weighted aggregation (WMMA f32), head-mean folded in
  gat_agg_wmma<<<dim3(N_NODES / 16, N_HID / 16), 32, 0, stream>>>(att, g, hid);

  // 4) column-normalize + clip a_mean in place
  gat_norm<<<(N_NODES * N_NODES) / 256, 256, 0, stream>>>(a_mean, rowmax);
}